// SmallScaleDeformation_19928648253745
// MI455X (gfx1250) — compile-verified
//
#include <hip/hip_runtime.h>
#include <stdint.h>

typedef __attribute__((ext_vector_type(4))) unsigned int u32x4;
typedef __attribute__((ext_vector_type(4))) int i32x4;
typedef __attribute__((ext_vector_type(8))) int i32x8;

#define B_ 16
#define C_ 31
#define H_ 256
#define W_ 256
#define R_ 25
#define TILE_ 64
#define ST_ (TILE_ + 2 * R_)   // 114 -> 114*114*4 = 51984 B of LDS

__device__ __forceinline__ int imax(int a, int b) { return a > b ? a : b; }
__device__ __forceinline__ int imin(int a, int b) { return a < b ? a : b; }

// Threefry2x32, 20 rounds, JAX rotation schedule.
__device__ __forceinline__ void threefry2x32(uint32_t k0, uint32_t k1,
                                             uint32_t x0, uint32_t x1,
                                             uint32_t& o0, uint32_t& o1) {
  const uint32_t k2 = k0 ^ k1 ^ 0x1BD11BDAu;
  uint32_t y0 = x0 + k0, y1 = x1 + k1;
#define TFR(rot) { y0 += y1; y1 = (y1 << rot) | (y1 >> (32 - rot)); y1 ^= y0; }
  TFR(13) TFR(15) TFR(26) TFR(6)
  y0 += k1; y1 += k2 + 1u;
  TFR(17) TFR(29) TFR(16) TFR(24)
  y0 += k2; y1 += k0 + 2u;
  TFR(13) TFR(15) TFR(26) TFR(6)
  y0 += k0; y1 += k1 + 3u;
  TFR(17) TFR(29) TFR(16) TFR(24)
  y0 += k1; y1 += k2 + 4u;
  TFR(13) TFR(15) TFR(26) TFR(6)
  y0 += k2; y1 += k0 + 5u;
#undef TFR
  o0 = y0; o1 = y1;
}

// Decode one 32-bit random word -> LDS element index of the deformed source pixel.
__device__ __forceinline__ uint32_t lds_offset(uint32_t u, int r, int c,
                                               int stW, int r_lo, int c_lo) {
  // P(deform) = 51/256 ~= 0.2 ; shift in [-25, +25] via %51
  int sx = ((u & 0xFFu) < 51u) ? ((int)((u >> 8) & 0xFFu) % 51) - R_ : 0;
  int sy = (((u >> 16) & 0xFFu) < 51u) ? ((int)((u >> 24) & 0xFFu) % 51) - R_ : 0;
  int sc = imin(W_ - 1, imax(0, c + sx));
  int sr = imin(H_ - 1, imax(0, r + sy));
  return (uint32_t)((sr - r_lo) * stW + (sc - c_lo));
}

__global__ __launch_bounds__(256, 1)
void SmallScaleDeformation_19928648253745_kernel(const float* __restrict__ in,
                                                 float* __restrict__ out) {
  __shared__ uint32_t sh[ST_ * ST_];

  const int tilesPerSide  = H_ / TILE_;                  // 4
  const int tilesPerPlane = tilesPerSide * tilesPerSide; // 16
  const int plane = blockIdx.x / tilesPerPlane;          // b*C + c
  const int t     = blockIdx.x % tilesPerPlane;
  const int r0 = (t / tilesPerSide) * TILE_;
  const int c0 = (t % tilesPerSide) * TILE_;

  const int r_lo = imax(0, r0 - R_);
  const int r_hi = imin(H_, r0 + TILE_ + R_);
  const int c_lo = imax(0, c0 - R_);
  const int c_hi = imin(W_, c0 + TILE_ + R_);
  const int stRows = r_hi - r_lo;
  const int stW    = c_hi - c_lo;

  const int tid = threadIdx.x;

#if __has_builtin(__builtin_amdgcn_tensor_load_to_lds)
  // ---- Phase 1: one wave kicks off the TDM 2D tile DMA (global strided -> LDS packed).
  if (tid < 32) {
    const uint64_t ga  = (uint64_t)(uintptr_t)(in + ((size_t)plane * H_ + r_lo) * W_ + c_lo);
    const uint32_t lds = (uint32_t)(uintptr_t)(&sh[0]);

    // D# group 0 (128b): count=1 | lds_addr | global_addr[56:0] | type=2
    uint32_t d0 = 1u;
    uint32_t d1 = lds;
    uint32_t d2 = (uint32_t)ga;
    uint32_t d3 = ((uint32_t)(ga >> 32) & 0x01FFFFFFu) | (2u << 30);

    // D# group 1 (256b): data_size=4B, 2D dims/tile, row pitch = W
    uint32_t e0 = (2u << 16);
    uint32_t e1 = ((uint32_t)stW & 0xFFFFu) << 16;
    uint32_t e2 = ((uint32_t)stRows & 0xFFFFu) << 16;
    uint32_t e3 = ((uint32_t)stW) << 16;
    uint32_t e4 = (uint32_t)stRows;
    uint32_t e5 = (uint32_t)W_;
    uint32_t e6 = 0u;
    uint32_t e7 = 0u;

    u32x4 g0 = { (unsigned)__builtin_amdgcn_readfirstlane((int)d0),
                 (unsigned)__builtin_amdgcn_readfirstlane((int)d1),
                 (unsigned)__builtin_amdgcn_readfirstlane((int)d2),
                 (unsigned)__builtin_amdgcn_readfirstlane((int)d3) };
    i32x8 g1 = { __builtin_amdgcn_readfirstlane((int)e0),
                 __builtin_amdgcn_readfirstlane((int)e1),
                 __builtin_amdgcn_readfirstlane((int)e2),
                 __builtin_amdgcn_readfirstlane((int)e3),
                 __builtin_amdgcn_readfirstlane((int)e4),
                 __builtin_amdgcn_readfirstlane((int)e5),
                 __builtin_amdgcn_readfirstlane((int)e6),
                 __builtin_amdgcn_readfirstlane((int)e7) };
    i32x4 z4 = { 0, 0, 0, 0 };
#if __clang_major__ >= 23
    i32x8 z8 = { 0, 0, 0, 0, 0, 0, 0, 0 };
    __builtin_amdgcn_tensor_load_to_lds(g0, g1, z4, z4, z8, 0);
#else
    __builtin_amdgcn_tensor_load_to_lds(g0, g1, z4, z4, 0);
#endif
  }
#endif

  // ---- Phase 2 (overlapped with the DMA): threefry RNG + decode into LDS offsets.
  // Pixel k (k=0..15) of this thread is linear index p = k*256 + tid in the 64x64 tile.
  uint32_t offs[16];
#pragma unroll
  for (int it = 0; it < 8; ++it) {
    const int pA = it * 512 + tid;         // k = 2*it
    const int pB = pA + 256;               // k = 2*it+1
    const int rA = r0 + (pA >> 6), cA = c0 + (pA & 63);
    const int rB = r0 + (pB >> 6), cB = c0 + (pB & 63);
    const uint32_t idxA = (uint32_t)(plane * (H_ * W_) + rA * W_ + cA);
    uint32_t u0, u1;
    threefry2x32(42u, 0xCAFEF00Du, idxA, 0x9E3779B9u, u0, u1);
    offs[2 * it]     = lds_offset(u0, rA, cA, stW, r_lo, c_lo);
    offs[2 * it + 1] = lds_offset(u1, rB, cB, stW, r_lo, c_lo);
  }

#if __has_builtin(__builtin_amdgcn_tensor_load_to_lds)
  if (tid < 32) {
    __builtin_amdgcn_s_wait_tensorcnt(0);
  }
#else
  // Fallback: cooperative staged copy if the TDM builtin is unavailable.
  for (int i = tid; i < stRows * stW; i += 256) {
    int rr = i / stW, cc = i % stW;
    sh[i] = __float_as_uint(in[((size_t)plane * H_ + r_lo + rr) * W_ + c_lo + cc]);
  }
#endif
  __syncthreads();

  // ---- Phase 3: 16 LDS gathers + 16 coalesced non-temporal stores (write-once data;
  // keep L2 capacity for the input halo re-reads).
  float* const outPlane = out + (size_t)plane * (H_ * W_);
#pragma unroll
  for (int k = 0; k < 16; ++k) {
    const int p = k * 256 + tid;
    const int r = r0 + (p >> 6), c = c0 + (p & 63);
    const float v = __uint_as_float(sh[offs[k]]);
    __builtin_nontemporal_store(v, outPlane + r * W_ + c);
  }
}

extern "C" void kernel_launch(void* const* d_in, const int* in_sizes, int n_in,
                              void* d_out, int out_size, void* d_ws, size_t ws_size,
                              hipStream_t stream) {
  const float* in = (const float*)d_in[0];
  float* out = (float*)d_out;
  const int blocks = B_ * C_ * (H_ / TILE_) * (W_ / TILE_); // 16*31*4*4 = 7936
  hipLaunchKernelGGL(SmallScaleDeformation_19928648253745_kernel,
                     dim3(blocks), dim3(256), 0, stream, in, out);
}